// LabelEncoder_35820027249126
// MI455X (gfx1250) — compile-verified
//
#include <hip/hip_runtime.h>

// out[b, :] = wrap(W[labels[b], :] + noise[b, :]),  wrap(x) = mod(x+1,2)-1
// Memory-bound gather: ~320 MB HBM traffic => ~14us floor @ 23.3 TB/s.
// Strategy: b128 per lane, NT hints on streamed tensors (keep 51MB W table
// resident in the 192MB L2), gfx1250 global_prefetch_b8 to hide gather latency.

typedef float v4f __attribute__((ext_vector_type(4)));

#define N_D      128
#define F4_PER_ROW 32   // 128 floats / 4

__global__ __launch_bounds__(256)
void label_encoder_gather_kernel(const int*   __restrict__ labels,
                                 const float* __restrict__ W,
                                 const float* __restrict__ noise,
                                 float*       __restrict__ out,
                                 int total4,     // nrows * 32 float4 elements
                                 int stride4)    // gridDim.x * blockDim.x
{
    int g = blockIdx.x * blockDim.x + threadIdx.x;

    for (; g < total4; g += stride4) {
        const int row  = g >> 5;        // which batch row (32 float4 per row)
        const int col4 = g & (F4_PER_ROW - 1);
        const int label = labels[row];

        // Prefetch next iteration's gathered W row: the label is known well
        // before the data is needed, so issue a speculative prefetch
        // (lowers to global_prefetch_b8 on gfx1250). Locality=3: keep in L2,
        // W is the hot, reused table.
        const int gn = g + stride4;
        if (gn < total4) {
            const int labn = labels[gn >> 5];
            __builtin_prefetch(&W[(long)labn * N_D + ((gn & (F4_PER_ROW - 1)) << 2)], 0, 3);
        }

        // Gather from W: regular-temporal load -> W becomes L2-resident.
        const v4f w = *(const v4f*)&W[(long)label * N_D + (col4 << 2)];

        // Streaming noise: read exactly once -> non-temporal so it doesn't
        // evict the W table from L2.
        const v4f n = __builtin_nontemporal_load((const v4f*)&noise[(long)g * 4]);

        const v4f x = w + n;

        // wrap into [-1,1): r = x - 2*floor((x+1)/2)   (3 VALU ops / element)
        v4f r;
#pragma unroll
        for (int k = 0; k < 4; ++k) {
            const float f = __builtin_floorf(__builtin_fmaf(x[k], 0.5f, 0.5f));
            r[k] = __builtin_fmaf(-2.0f, f, x[k]);
        }

        // Output written once: non-temporal b128 store.
        __builtin_nontemporal_store(r, (v4f*)&out[(long)g * 4]);
    }
}

extern "C" void kernel_launch(void* const* d_in, const int* in_sizes, int n_in,
                              void* d_out, int out_size, void* d_ws, size_t ws_size,
                              hipStream_t stream) {
    const int*   labels = (const int*)  d_in[0];   // [BATCH]
    const float* W      = (const float*)d_in[1];   // [N_CLASSES, 128]
    const float* noise  = (const float*)d_in[2];   // [BATCH, 128]
    float*       out    = (float*)d_out;           // [BATCH, 128]

    const int nrows  = in_sizes[0];
    const int total4 = nrows * F4_PER_ROW;         // float4 elements to produce

    const int threads = 256;                       // 8 wave32s per block
    int blocks = (total4 + threads - 1) / threads;
    if (blocks > 4096) blocks = 4096;              // grid-stride fills the rest
    const int stride4 = blocks * threads;

    label_encoder_gather_kernel<<<blocks, threads, 0, stream>>>(
        labels, W, noise, out, total4, stride4);
}